// NodeAttention_25606595019108
// MI455X (gfx1250) — compile-verified
//
#include <hip/hip_runtime.h>
#include <math.h>

// ---------------------------------------------------------------------------
// GAT (DGL GATConv-style) forward for MI455X / gfx1250.
// N=50000 nodes, E=800000 edges, IN=128, HEADS=4, OUT=32 (H*D = 128).
// GEMM uses V_WMMA_F32_16X16X4_F32 (fp32 matrix core path, matches fp32 ref).
// ---------------------------------------------------------------------------

typedef float v2f __attribute__((ext_vector_type(2)));
typedef float v8f __attribute__((ext_vector_type(8)));

#define HEADS 4
#define OUTF 32
#define FEATW 128   // HEADS*OUTF == IN_FEAT == 128
#define NEG_SLOPE 0.2f

// order-preserving float <-> uint key (for exact atomic float max)
__device__ __forceinline__ unsigned f2key(float f) {
  unsigned b = __float_as_uint(f);
  return (b & 0x80000000u) ? ~b : (b | 0x80000000u);
}
__device__ __forceinline__ float key2f(unsigned k) {
  unsigned b = (k & 0x80000000u) ? (k & 0x7FFFFFFFu) : ~k;
  return __uint_as_float(b);
}
#define NEG_INF_KEY 0x007FFFFFu   // f2key(-inf)

// ---------------------------------------------------------------------------
// 0) init: zero accumulator (d_out), zero denom, seg-max keys = key(-inf)
// ---------------------------------------------------------------------------
__global__ void gat_init(float* __restrict__ out, float* __restrict__ denom,
                         unsigned* __restrict__ key, int NF, int NH) {
  int i = blockIdx.x * blockDim.x + threadIdx.x;
  if (i < NF) out[i] = 0.0f;
  if (i < NH) { denom[i] = 0.0f; key[i] = NEG_INF_KEY; }
}

// ---------------------------------------------------------------------------
// 1) feat = h @ W via v_wmma_f32_16x16x4_f32.
//    One wave per 16x16 output tile; 8 waves (8 nTiles, one mTile) per block.
//    N is a multiple of 16 (50000 = 3125*16) so no edge guards needed.
// ---------------------------------------------------------------------------
__global__ __launch_bounds__(256) void gat_gemm_wmma(
    const float* __restrict__ h, const float* __restrict__ W,
    float* __restrict__ feat) {
  const int wave  = threadIdx.x >> 5;
  const int lane  = threadIdx.x & 31;
  const int gw    = blockIdx.x * 8 + wave;
  const int mTile = gw >> 3;          // 0 .. N/16-1
  const int nTile = gw & 7;           // 0 .. 7  (128/16)
  const int half  = lane >> 4;        // 0 or 1
  const int lr    = lane & 15;

  const int row = mTile * 16 + lr;    // A row read by this lane
  const int col = nTile * 16 + lr;    // B col / D col for this lane
  const float* __restrict__ arow = h + (size_t)row * FEATW;

  v8f c = {};
#pragma unroll 4
  for (int k0 = 0; k0 < FEATW; k0 += 4) {
    const int ka = k0 + 2 * half;     // K index pair handled by this lane-half
    v2f a, b;
    a.x = arow[ka];
    a.y = arow[ka + 1];
    b.x = W[(size_t)ka * FEATW + col];
    b.y = W[(size_t)(ka + 1) * FEATW + col];
    // D = A(16x4) * B(4x16) + C ;  emits v_wmma_f32_16x16x4_f32
    c = __builtin_amdgcn_wmma_f32_16x16x4_f32(
        /*neg_a=*/false, a, /*neg_b=*/false, b,
        /*c_mod=*/(short)0, c, /*reuse_a=*/false, /*reuse_b=*/false);
  }

  const int mbase = mTile * 16 + half * 8;   // C/D: VGPR j -> M = j + 8*half
#pragma unroll
  for (int j = 0; j < 8; ++j) {
    feat[(size_t)(mbase + j) * FEATW + col] = c[j];
  }
}

// ---------------------------------------------------------------------------
// 2) el[n,h] = <feat[n,h,:], attn_l[h,:]> ; er likewise. One thread per (n,h).
// ---------------------------------------------------------------------------
__global__ void gat_logits(const float* __restrict__ feat,
                           const float* __restrict__ attn_l,
                           const float* __restrict__ attn_r,
                           float* __restrict__ el, float* __restrict__ er, int NH) {
  int i = blockIdx.x * blockDim.x + threadIdx.x;
  if (i >= NH) return;
  const int head = i & (HEADS - 1);
  const float* __restrict__ f  = feat + (size_t)(i >> 2) * FEATW + head * OUTF;
  const float* __restrict__ al = attn_l + head * OUTF;
  const float* __restrict__ ar = attn_r + head * OUTF;
  float sl = 0.0f, sr = 0.0f;
#pragma unroll
  for (int d = 0; d < OUTF; ++d) {
    float v = f[d];
    sl = fmaf(v, al[d], sl);
    sr = fmaf(v, ar[d], sr);
  }
  el[i] = sl;
  er[i] = sr;
}

// ---------------------------------------------------------------------------
// 3) per-(edge,head) score + atomic segment max into key[dst,h]
// ---------------------------------------------------------------------------
__global__ void gat_edge_max(const int* __restrict__ src, const int* __restrict__ dst,
                             const float* __restrict__ el, const float* __restrict__ er,
                             unsigned* __restrict__ key, int EH) {
  int i = blockIdx.x * blockDim.x + threadIdx.x;
  if (i >= EH) return;
  const int e = i >> 2, hh = i & (HEADS - 1);
  const int s = src[e], d = dst[e];
  float x = el[s * HEADS + hh] + er[d * HEADS + hh];
  x = (x > 0.0f) ? x : NEG_SLOPE * x;           // leaky_relu
  atomicMax(&key[d * HEADS + hh], f2key(x));
}

// ---------------------------------------------------------------------------
// 4) decode keys -> seg_max, zero-in-degree guard (ref: where(isfinite, m, 0))
// ---------------------------------------------------------------------------
__global__ void gat_fix_max(const unsigned* __restrict__ key,
                            float* __restrict__ segmax, int NH) {
  int i = blockIdx.x * blockDim.x + threadIdx.x;
  if (i >= NH) return;
  float m = key2f(key[i]);
  segmax[i] = isfinite(m) ? m : 0.0f;
}

// ---------------------------------------------------------------------------
// 5) denom[dst,h] += exp(e - seg_max[dst,h])
// ---------------------------------------------------------------------------
__global__ void gat_edge_denom(const int* __restrict__ src, const int* __restrict__ dst,
                               const float* __restrict__ el, const float* __restrict__ er,
                               const float* __restrict__ segmax,
                               float* __restrict__ denom, int EH) {
  int i = blockIdx.x * blockDim.x + threadIdx.x;
  if (i >= EH) return;
  const int e = i >> 2, hh = i & (HEADS - 1);
  const int s = src[e], d = dst[e];
  const int idx = d * HEADS + hh;
  float x = el[s * HEADS + hh] + er[idx];
  x = (x > 0.0f) ? x : NEG_SLOPE * x;
  atomicAdd(&denom[idx], __expf(x - segmax[idx]));
}

// ---------------------------------------------------------------------------
// 6) out[dst,h,d] += alpha * feat[src,h,d]   (one block of 128 per edge;
//    thread t -> head = t>>5, feature = t&31; head-wide loads broadcast)
// ---------------------------------------------------------------------------
__global__ __launch_bounds__(128) void gat_scatter(
    const int* __restrict__ src, const int* __restrict__ dst,
    const float* __restrict__ el, const float* __restrict__ er,
    const float* __restrict__ segmax, const float* __restrict__ denom,
    const float* __restrict__ feat, float* __restrict__ out) {
  const int e = blockIdx.x;
  const int t = threadIdx.x;            // 0..127
  const int hh = t >> 5;
  const int s = src[e], d = dst[e];
  const int idx = d * HEADS + hh;
  float x = el[s * HEADS + hh] + er[idx];
  x = (x > 0.0f) ? x : NEG_SLOPE * x;
  const float alpha = __expf(x - segmax[idx]) / denom[idx];
  atomicAdd(&out[(size_t)d * FEATW + t], alpha * feat[(size_t)s * FEATW + t]);
}

// ---------------------------------------------------------------------------
// 7) out = elu(out + bias)  (in place; each thread owns one element)
// ---------------------------------------------------------------------------
__global__ void gat_final(float* __restrict__ out, const float* __restrict__ bias, int NF) {
  int i = blockIdx.x * blockDim.x + threadIdx.x;
  if (i >= NF) return;
  float v = out[i] + bias[i & (FEATW - 1)];
  out[i] = (v > 0.0f) ? v : (__expf(v) - 1.0f);
}

// ---------------------------------------------------------------------------
extern "C" void kernel_launch(void* const* d_in, const int* in_sizes, int n_in,
                              void* d_out, int out_size, void* d_ws, size_t ws_size,
                              hipStream_t stream) {
  const float* h      = (const float*)d_in[0];   // [N,128]
  const float* W      = (const float*)d_in[1];   // [128,128]
  const float* attn_l = (const float*)d_in[2];   // [4,32]
  const float* attn_r = (const float*)d_in[3];   // [4,32]
  const float* bias   = (const float*)d_in[4];   // [4,32]
  const int*   src    = (const int*)d_in[5];     // [E]
  const int*   dst    = (const int*)d_in[6];     // [E]

  const int N  = in_sizes[0] / FEATW;            // 50000
  const int E  = in_sizes[5];                    // 800000
  const int NH = N * HEADS;
  const int NF = N * FEATW;
  const int EH = E * HEADS;

  // workspace layout (floats): feat | el | er | segmax | denom | keys
  float*    feat   = (float*)d_ws;
  float*    el     = feat + (size_t)NF;
  float*    er     = el + NH;
  float*    segmax = er + NH;
  float*    denom  = segmax + NH;
  unsigned* key    = (unsigned*)(denom + NH);

  float* out = (float*)d_out;

  const int B = 256;
  gat_init<<<(NF + B - 1) / B, B, 0, stream>>>(out, denom, key, NF, NH);

  // GEMM: (N/16)*8 waves, 8 waves per block -> N/16 blocks
  gat_gemm_wmma<<<N / 16, 256, 0, stream>>>(h, W, feat);

  gat_logits<<<(NH + B - 1) / B, B, 0, stream>>>(feat, attn_l, attn_r, el, er, NH);

  gat_edge_max<<<(EH + B - 1) / B, B, 0, stream>>>(src, dst, el, er, key, EH);

  gat_fix_max<<<(NH + B - 1) / B, B, 0, stream>>>(key, segmax, NH);

  gat_edge_denom<<<(EH + B - 1) / B, B, 0, stream>>>(src, dst, el, er, segmax, denom, EH);

  gat_scatter<<<E, 128, 0, stream>>>(src, dst, el, er, segmax, denom, feat, out);

  gat_final<<<(NF + B - 1) / B, B, 0, stream>>>(out, bias, NF);
}